// MultiQubitMeasurement_28681791602942
// MI455X (gfx1250) — compile-verified
//
#include <hip/hip_runtime.h>
#include <math.h>

typedef float v2f __attribute__((ext_vector_type(2)));
typedef float v8f __attribute__((ext_vector_type(8)));

#define NQ 4
#define NL 3

// ---------------------------------------------------------------------------
// Setup kernel: build M[q][j][n] = Re( sum_k conj(V[k][j]) * z_q(k) * V[k][n] )
// where V is the 16x16 unitary of the variational layers (weights only).
// Runs once per launch, 1 block. Index convention: k = b0*8+b1*4+b2*2+b3,
// wire i <-> bit position (3-i), matching the reference's axis ordering.
// ---------------------------------------------------------------------------
struct c2 { float x, y; };
__device__ __forceinline__ c2 cmul(c2 a, c2 b) { return { a.x*b.x - a.y*b.y, a.x*b.y + a.y*b.x }; }
__device__ __forceinline__ c2 cadd(c2 a, c2 b) { return { a.x+b.x, a.y+b.y }; }

__global__ void build_M_kernel(const float* __restrict__ w, float* __restrict__ M)
{
    __shared__ c2 V[16][16];              // V[k][column]
    const int tid = threadIdx.x;
    if (tid < 16) {
        const int col = tid;
        for (int k = 0; k < 16; ++k) V[k][col] = { (k == col) ? 1.f : 0.f, 0.f };
        for (int l = 0; l < NL; ++l) {
            for (int i = 0; i < NQ; ++i) {
                for (int g = 0; g < 3; ++g) {          // RX, RY, RZ
                    float t = 0.5f * w[(l*NQ + i)*3 + g];
                    float c = cosf(t), s = sinf(t);
                    c2 U00, U01, U10, U11;
                    if (g == 0)      { U00 = {c,0};  U01 = {0,-s}; U10 = {0,-s}; U11 = {c,0}; } // RX
                    else if (g == 1) { U00 = {c,0};  U01 = {-s,0}; U10 = {s,0};  U11 = {c,0}; } // RY
                    else             { U00 = {c,-s}; U01 = {0,0};  U10 = {0,0};  U11 = {c,s}; } // RZ
                    const int p = 3 - i;
                    for (int k = 0; k < 16; ++k) {
                        if (k & (1 << p)) continue;
                        int k1 = k | (1 << p);
                        c2 a = V[k][col], b = V[k1][col];
                        V[k][col]  = cadd(cmul(U00, a), cmul(U01, b));
                        V[k1][col] = cadd(cmul(U10, a), cmul(U11, b));
                    }
                }
            }
            for (int i = 0; i < NQ; ++i) {             // CNOT(i, (i+1)%4), in order
                int pc = 3 - i, pt = 3 - ((i + 1) & 3);
                for (int k = 0; k < 16; ++k) {
                    if ((k & (1 << pc)) && !(k & (1 << pt))) {
                        int k1 = k | (1 << pt);
                        c2 tmp = V[k][col]; V[k][col] = V[k1][col]; V[k1][col] = tmp;
                    }
                }
            }
        }
    }
    __syncthreads();
    // 256 threads: thread -> (j, n); all four q at once.
    const int j = tid >> 4, n = tid & 15;
    float acc[4] = {0.f, 0.f, 0.f, 0.f};
    for (int k = 0; k < 16; ++k) {
        c2 a = V[k][j], b = V[k][n];
        float re = a.x*b.x + a.y*b.y;                  // Re(conj(a)*b)
        #pragma unroll
        for (int q = 0; q < 4; ++q)
            acc[q] += (((k >> (3 - q)) & 1) ? -re : re);
    }
    #pragma unroll
    for (int q = 0; q < 4; ++q) M[q*256 + j*16 + n] = acc[q];
}

// ---------------------------------------------------------------------------
// Main kernel: one wave handles a tile of 16 samples per loop iteration.
//   D_q = M_q (A, 16x16 via 4 chained 16x16x4 f32 WMMAs) x Psi^T (B)
//   <Z_q>[s] = sum_j psi[s][j] * D_q[j][s]  -> 8 FMAs + one shfl_xor(16)
// A-layout (f32 16x4): lane -> row M=lane&15, VGPR r -> K = 4t + 2*hi + r
// B-layout (f32 4x16): lane -> col N=lane&15, VGPR r -> K = 4t + 2*hi + r
// D-layout (ISA):      lane -> col N=lane&15, VGPR r -> row M = r + 8*hi
//
// psi[j], j = b0*8+b1*4+b2*2+b3:  psi = (b0?s0:c0)(b1?s1:c1)(b2?s0:c0)(b3?s1:c1)
// All hi-dependence is funneled through ONE select (a2 = hi ? s0 : c0) so the
// compiler never sees a dynamically-indexed register array (no cndmask chains).
// ---------------------------------------------------------------------------
__global__ void __launch_bounds__(256)
qnn_expval_kernel(const float* __restrict__ x, const float* __restrict__ M,
                  float* __restrict__ out, int n, int ntiles)
{
    const int lane = threadIdx.x & 31;
    const int nl   = lane & 15;
    const int hi   = lane >> 4;
    const int wave   = (blockIdx.x * blockDim.x + threadIdx.x) >> 5;
    const int nwaves = (gridDim.x * blockDim.x) >> 5;

    // Tile-invariant A operands: Areg[q][t][r] = M_q[nl][4t + 2*hi + r]
    v2f Areg[4][4];
    #pragma unroll
    for (int q = 0; q < 4; ++q)
        #pragma unroll
        for (int t = 0; t < 4; ++t) {
            int j0 = 4*t + 2*hi;
            Areg[q][t][0] = M[q*256 + nl*16 + j0 + 0];
            Areg[q][t][1] = M[q*256 + nl*16 + j0 + 1];
        }

    for (int tile = wave; tile < ntiles; tile += nwaves) {
        const int base = tile * 16;
        const int s    = base + nl;                 // sample owned by this lane
        const int sc   = (s < n) ? s : (n - 1);     // clamp for partial tail tile
        float2 xv = reinterpret_cast<const float2*>(x)[sc];

        float c0, s0, c1, s1;
        sincosf(0.5f * xv.x, &s0, &c0);
        sincosf(0.5f * xv.y, &s1, &c1);

        // Single runtime select; everything else is compile-time factor picks.
        const float a2  = hi ? s0 : c0;             // wire-2 factor for this half
        const float m0  = c0 * a2;                  // (b0=0, b2=hi)
        const float m1  = s0 * a2;                  // (b0=1, b2=hi)
        const float w00 = c1 * c1;                  // (b1,b3) = (0,0)
        const float w01 = c1 * s1;                  // (0,1) == (1,0)
        const float w11 = s1 * s1;                  // (1,1)

        v8f acc[4];
        #pragma unroll
        for (int q = 0; q < 4; ++q)
            #pragma unroll
            for (int e = 0; e < 8; ++e) acc[q][e] = 0.f;

        #pragma unroll
        for (int t = 0; t < 4; ++t) {
            // K = 4t + 2*hi + r  ->  bits (b0,b1,b2,b3) = (t>=2, t&1, hi, r)
            const float bt = (t >= 2) ? m1 : m0;    // compile-time pick
            v2f b;
            b[0] = bt * ((t & 1) ? w01 : w00);      // r = 0  (b3 = 0)
            b[1] = bt * ((t & 1) ? w11 : w01);      // r = 1  (b3 = 1)
            #pragma unroll
            for (int q = 0; q < 4; ++q)
                acc[q] = __builtin_amdgcn_wmma_f32_16x16x4_f32(
                             false, Areg[q][t], false, b,
                             (short)0, acc[q], false, false);
        }

        // pr[r] = psi[r + 8*hi] : bits (b0,b1,b2,b3) = (hi, r>=4, r&2, r&1)
        float pr[8];
        #pragma unroll
        for (int r = 0; r < 8; ++r) {
            const float u = (r & 2) ? m1 : m0;
            const float v = (r & 4) ? ((r & 1) ? w11 : w01)
                                    : ((r & 1) ? w01 : w00);
            pr[r] = u * v;
        }

        float o[4];
        #pragma unroll
        for (int q = 0; q < 4; ++q) {
            float p = 0.f;
            #pragma unroll
            for (int r = 0; r < 8; ++r) p = fmaf(pr[r], acc[q][r], p);
            p += __shfl_xor(p, 16, 32);             // combine the two halves
            o[q] = p;
        }

        if (hi == 0 && s < n)                       // coalesced float4 per sample
            reinterpret_cast<float4*>(out)[s] = make_float4(o[0], o[1], o[2], o[3]);
    }
}

// ---------------------------------------------------------------------------
extern "C" void kernel_launch(void* const* d_in, const int* in_sizes, int n_in,
                              void* d_out, int out_size, void* d_ws, size_t ws_size,
                              hipStream_t stream)
{
    const float* x = (const float*)d_in[0];   // (B, 2) fp32
    const float* w = (const float*)d_in[1];   // (3, 4, 3) fp32
    float* out = (float*)d_out;               // (B, 4) fp32
    float* M   = (float*)d_ws;                // 4 * 16 * 16 floats = 4 KB

    const int n      = in_sizes[0] / 2;
    const int ntiles = (n + 15) / 16;

    build_M_kernel<<<1, 256, 0, stream>>>(w, M);

    const int waves_per_block = 256 / 32;
    int blocks = (ntiles + waves_per_block - 1) / waves_per_block;
    if (blocks > 2048) blocks = 2048;         // grid-stride: ~16K waves, ~4 tiles each
    qnn_expval_kernel<<<blocks, 256, 0, stream>>>(x, M, out, n, ntiles);
}